// SpikingNeuron_17145509446145
// MI455X (gfx1250) — compile-verified
//
#include <hip/hip_runtime.h>
#include <stdint.h>

// Problem shape (fixed by the reference)
#define T_STEPS 512
#define F_NEUR  65536
// Persistent-kernel shape: 64 blocks * 256 threads * 4 floats = 65536 neurons
#define NBLOCKS  64
#define NTHREADS 256
#define VEC      4

#define DECAY   0.9f
#define THRESH  1.0f
#define AMP     1.0f
#define REFRAC  2

// Native clang vector types (HIP's float4 is a struct; builtins need these).
typedef float v4f __attribute__((ext_vector_type(4)));
typedef __attribute__((__vector_size__(4 * sizeof(int)))) int v4i_b128;

// ---------------------------------------------------------------------------
// CDNA5 async global->LDS copy (tracked by ASYNCcnt) + wait.
// ---------------------------------------------------------------------------
__device__ __forceinline__ void async_f4_to_lds(const float* gsrc, v4f* lds_dst) {
#if defined(__gfx1250__)
#if __has_builtin(__builtin_amdgcn_global_load_async_to_lds_b128)
  __builtin_amdgcn_global_load_async_to_lds_b128(
      (__attribute__((address_space(1))) v4i_b128*)(uintptr_t)gsrc,
      (__attribute__((address_space(3))) v4i_b128*)(uint32_t)(uintptr_t)lds_dst,
      0, 0);
#else
  uint32_t lds_addr = (uint32_t)(uintptr_t)lds_dst;
  asm volatile("global_load_async_to_lds_b128 %0, %1, off"
               :
               : "v"(lds_addr), "v"(gsrc)
               : "memory");
#endif
#else
  *lds_dst = *(const v4f*)gsrc;      // host / non-gfx1250 parse path
#endif
}

__device__ __forceinline__ void wait_async0() {
#if defined(__gfx1250__)
#if __has_builtin(__builtin_amdgcn_s_wait_asynccnt)
  __builtin_amdgcn_s_wait_asynccnt(0);
#else
  asm volatile("s_wait_asynccnt 0x0" ::: "memory");
#endif
#endif
}

// ---------------------------------------------------------------------------
// Grid-wide split barrier for step t: post block any-bit, spin until all 64
// blocks arrived, return the global any-spike bit (uniform across block).
// ---------------------------------------------------------------------------
__device__ __forceinline__ uint32_t grid_sync_any(int t, bool anyLane,
                                                  uint32_t* flags, uint32_t* arrive,
                                                  uint32_t& s_any, uint32_t& s_flag) {
  const int tid = threadIdx.x;
  if (((tid & 31) == 0) && __any(anyLane)) atomicOr(&s_any, 1u);
  __syncthreads();
  if (tid == 0) {
    if (s_any)
      __hip_atomic_fetch_or(&flags[t], 1u, __ATOMIC_RELEASE,
                            __HIP_MEMORY_SCOPE_AGENT);
    __hip_atomic_fetch_add(&arrive[t], 1u, __ATOMIC_ACQ_REL,
                           __HIP_MEMORY_SCOPE_AGENT);
    while (__hip_atomic_load(&arrive[t], __ATOMIC_ACQUIRE,
                             __HIP_MEMORY_SCOPE_AGENT) < (uint32_t)NBLOCKS)
      __builtin_amdgcn_s_sleep(1);
    s_flag = __hip_atomic_load(&flags[t], __ATOMIC_ACQUIRE,
                               __HIP_MEMORY_SCOPE_AGENT);
  }
  __syncthreads();
  return s_flag;
}

// ---------------------------------------------------------------------------
// Zero the cross-block sync area in d_ws each launch (d_ws is poisoned 0xAA).
// ---------------------------------------------------------------------------
__global__ void lif_zero_ws(uint32_t* p, int n) {
  int i = blockIdx.x * blockDim.x + threadIdx.x;
  if (i < n) p[i] = 0u;
}

// ---------------------------------------------------------------------------
// Persistent LIF scan. Membrane state lives in VGPRs for all 512 steps.
//
// Key structure:
//  - Integration steps (r==0): consume x_t from LDS (async-to-LDS,
//    ASYNCcnt-waited), update/threshold/reset, grid-wide any() barrier,
//    then prefetch the *next integration step's* tile (t+r+1) so the HBM
//    round trip overlaps the refractory steps and the next spin.
//  - Refractory steps with thr>=0: provably spike-free (v<=thr and gate=0
//    imply v*decay<=thr), so NO grid sync and NO input load: pure decay.
//    ~2/3 of steps take this path -> ~3x fewer grid barriers, ~85MB of
//    input never read.
//  - thr<0 fallback: spikes possible while refractory; sync every step.
// ---------------------------------------------------------------------------
__global__ void __launch_bounds__(NTHREADS, 1)
lif_scan_kernel(const float* __restrict__ x,
                const float* __restrict__ wscale_p,
                const float* __restrict__ thr_adapt_p,
                float* __restrict__ out,
                uint32_t* __restrict__ flags,    // [T] any-spike flag per step
                uint32_t* __restrict__ arrive)   // [T] block arrival counters
{
  __shared__ v4f      xbuf[NTHREADS];      // single async tile buffer (4 KB)
  __shared__ uint32_t s_any;               // block-local any-spike
  __shared__ uint32_t s_flag;              // broadcast of global flag

  const int tid = threadIdx.x;
  const int col = (blockIdx.x * NTHREADS + tid) * VEC;

  const float wscale   = wscale_p[0];
  const float thr      = THRESH + thr_adapt_p[0];
  const bool  fastpath = (thr >= 0.0f);    // refractory steps provably silent

  v4f v = {0.f, 0.f, 0.f, 0.f};
  int r = 0;

  float* spikes = out;                                  // [T,F]
  float* vout   = out + (size_t)T_STEPS * F_NEUR;       // [T,F]
  const v4f zero4 = {0.f, 0.f, 0.f, 0.f};

  // Prime: t=0 is an integration step.
  async_f4_to_lds(x + col, &xbuf[tid]);

  for (int t = 0; t < T_STEPS; ++t) {
    v4f* sp_out = reinterpret_cast<v4f*>(spikes + (size_t)t * F_NEUR + col);
    v4f* v_out  = reinterpret_cast<v4f*>(vout   + (size_t)t * F_NEUR + col);

    if (r == 0) {
      // ---------------- integration step (always grid-synced) --------------
      if (tid == 0) s_any = 0u;
      __syncthreads();                     // reset visible; prior s_flag done

      wait_async0();
      v4f xv = xbuf[tid];

      v.x = v.x * DECAY + xv.x * wscale;
      v.y = v.y * DECAY + xv.y * wscale;
      v.z = v.z * DECAY + xv.z * wscale;
      v.w = v.w * DECAY + xv.w * wscale;

      const bool m0 = v.x > thr, m1 = v.y > thr, m2 = v.z > thr, m3 = v.w > thr;
      v4f sp = {m0 ? AMP : 0.f, m1 ? AMP : 0.f, m2 ? AMP : 0.f, m3 ? AMP : 0.f};
      if (m0) v.x = 0.f;
      if (m1) v.y = 0.f;
      if (m2) v.z = 0.f;
      if (m3) v.w = 0.f;

      __builtin_nontemporal_store(sp, sp_out);   // streamed once, bypass L2
      __builtin_nontemporal_store(v,  v_out);

      const uint32_t anyG =
          grid_sync_any(t, m0 | m1 | m2 | m3, flags, arrive, s_any, s_flag);
      r = anyG ? REFRAC : 0;

      // Prefetch the next tile we will actually consume.
      const int nxt = fastpath ? (t + r + 1) : (r == 0 ? t + 1 : -1);
      if (nxt >= 0 && nxt < T_STEPS)
        async_f4_to_lds(x + (size_t)nxt * F_NEUR + col, &xbuf[tid]);

    } else if (fastpath) {
      // ------- refractory, thr>=0: provably silent; no sync, no load -------
      v.x *= DECAY; v.y *= DECAY; v.z *= DECAY; v.w *= DECAY;
      __builtin_nontemporal_store(zero4, sp_out);
      __builtin_nontemporal_store(v,     v_out);
      --r;

    } else {
      // ------- refractory, thr<0: spikes possible; sync, but no load -------
      if (tid == 0) s_any = 0u;
      __syncthreads();

      v.x *= DECAY; v.y *= DECAY; v.z *= DECAY; v.w *= DECAY;
      const bool m0 = v.x > thr, m1 = v.y > thr, m2 = v.z > thr, m3 = v.w > thr;
      v4f sp = {m0 ? AMP : 0.f, m1 ? AMP : 0.f, m2 ? AMP : 0.f, m3 ? AMP : 0.f};
      if (m0) v.x = 0.f;
      if (m1) v.y = 0.f;
      if (m2) v.z = 0.f;
      if (m3) v.w = 0.f;

      __builtin_nontemporal_store(sp, sp_out);
      __builtin_nontemporal_store(v,  v_out);

      const uint32_t anyG =
          grid_sync_any(t, m0 | m1 | m2 | m3, flags, arrive, s_any, s_flag);
      r = anyG ? REFRAC : (r - 1);
      if (r == 0 && t + 1 < T_STEPS)        // next step integrates: fetch it
        async_f4_to_lds(x + (size_t)(t + 1) * F_NEUR + col, &xbuf[tid]);
    }
  }
}

// ---------------------------------------------------------------------------
extern "C" void kernel_launch(void* const* d_in, const int* in_sizes, int n_in,
                              void* d_out, int out_size, void* d_ws, size_t ws_size,
                              hipStream_t stream) {
  (void)in_sizes; (void)n_in; (void)out_size; (void)ws_size;

  const float* x      = (const float*)d_in[0];  // [T,F] f32
  const float* wscale = (const float*)d_in[1];  // scalar
  const float* tadapt = (const float*)d_in[2];  // scalar
  float*       out    = (float*)d_out;          // [2,T,F] f32 (spikes, v)

  uint32_t* flags  = (uint32_t*)d_ws;           // [T]
  uint32_t* arrive = flags + T_STEPS;           // [T]

  lif_zero_ws<<<(2 * T_STEPS + 255) / 256, 256, 0, stream>>>(flags, 2 * T_STEPS);
  lif_scan_kernel<<<NBLOCKS, NTHREADS, 0, stream>>>(x, wscale, tadapt, out,
                                                    flags, arrive);
}